// EnhancedAttention_70068096467384
// MI455X (gfx1250) — compile-verified
//
#include <hip/hip_runtime.h>

#define B_   2
#define S_   1024
#define H_   1024
#define NH_  16
#define HD_  64
#define SP_  0.05f
#define EPS_ 1e-5f

typedef float v2f __attribute__((ext_vector_type(2)));
typedef float v8f __attribute__((ext_vector_type(8)));

__device__ __forceinline__ v8f wmma4(v2f a, v2f b, v8f c) {
  // V_WMMA_F32_16X16X4_F32 : D = A(16x4) x B(4x16) + C(16x16), fp32 exact
  return __builtin_amdgcn_wmma_f32_16x16x4_f32(false, a, false, b, (short)0, c, false, false);
}

// CDNA5 async global->LDS copy (16B per lane), tracked by ASYNCcnt.
__device__ __forceinline__ void async_load_b128(void* lds_ptr, const void* gptr) {
  const unsigned lds_off = (unsigned)(unsigned long long)lds_ptr;  // low 32 bits = LDS byte offset
  asm volatile("global_load_async_to_lds_b128 %0, %1, off" :: "v"(lds_off), "v"(gptr) : "memory");
}
__device__ __forceinline__ void wait_async0() {
  asm volatile("s_wait_asynccnt 0x0" ::: "memory");
}

// ---------------- small reduction kernels ----------------

__global__ __launch_bounds__(256) void msb_mean_kernel(const float* __restrict__ msb,
                                                       float* __restrict__ msbh) {
  __shared__ float red[8];
  const int h = blockIdx.x;
  const float* p = msb + h * (HD_ * HD_);
  float s = 0.f;
  for (int i = threadIdx.x; i < HD_ * HD_; i += 256) s += p[i];
  for (int m = 16; m >= 1; m >>= 1) s += __shfl_xor(s, m, 32);
  const int lane = threadIdx.x & 31, wave = threadIdx.x >> 5;
  if (lane == 0) red[wave] = s;
  __syncthreads();
  if (threadIdx.x == 0) {
    float t = 0.f;
    for (int w = 0; w < 8; ++w) t += red[w];
    msbh[h] = t / (float)(HD_ * HD_);
  }
}

__global__ __launch_bounds__(256) void spec_in_kernel(const float* __restrict__ query,
                                                      float* __restrict__ specin) {
  const int g = blockIdx.x * 256 + threadIdx.x;   // 0 .. B*H-1
  const int b = g >> 10, c = g & (H_ - 1);
  const float* p = query + (size_t)b * S_ * H_ + c;
  float s = 0.f;
  for (int i = 0; i < S_; ++i) s += p[(size_t)i * H_];
  specin[g] = s / (float)S_;
}

__global__ __launch_bounds__(256) void spec_mlp_kernel(
    const float* __restrict__ specin,
    const float* __restrict__ Ws1, const float* __restrict__ bs1,
    const float* __restrict__ Ws2, const float* __restrict__ bs2,
    float* __restrict__ spec) {
  __shared__ float xin[H_];
  __shared__ float h1[H_ / 2];
  __shared__ float red[8];
  const int b = blockIdx.x;
  for (int i = threadIdx.x; i < H_; i += 256) xin[i] = specin[b * H_ + i];
  __syncthreads();
  for (int jj = 0; jj < 2; ++jj) {
    const int j = threadIdx.x * 2 + jj;
    float d = bs1[j];
    for (int k2 = 0; k2 < H_; ++k2) d += xin[k2] * Ws1[(size_t)k2 * (H_ / 2) + j];
    h1[j] = d > 0.f ? d : 0.f;
  }
  __syncthreads();
  float acc = 0.f;
  for (int nn = 0; nn < 4; ++nn) {
    const int n = threadIdx.x * 4 + nn;
    float d = bs2[n];
    for (int k2 = 0; k2 < H_ / 2; ++k2) d += h1[k2] * Ws2[(size_t)k2 * H_ + n];
    acc += 1.f / (1.f + __expf(-d));
  }
  for (int m = 16; m >= 1; m >>= 1) acc += __shfl_xor(acc, m, 32);
  const int lane = threadIdx.x & 31, wave = threadIdx.x >> 5;
  if (lane == 0) red[wave] = acc;
  __syncthreads();
  if (threadIdx.x == 0) {
    float t = 0.f;
    for (int w = 0; w < 8; ++w) t += red[w];
    spec[b] = t / (float)H_;
  }
}

// ---------------- double-buffered WMMA GEMM core (shared by QKV and O-proj) ----------------
// Block 256 thr = 8 waves; block tile 64(M) x 128(N); k-chunk 16.
// As padded to stride 20 (16 banks, float4-aligned); BsT transposed, stride 18 (b64 frags).

#define AS_STRIDE 20
#define BS_STRIDE 18

struct GemmAcc { v8f acc[4]; };

__device__ __forceinline__ void gemm_mainloop(
    const float* __restrict__ X, const float* __restrict__ W,
    int m0, int n0, float (*As)[64][AS_STRIDE], float (*BsT)[128][BS_STRIDE],
    GemmAcc& g) {
  const int tid = threadIdx.x, lane = tid & 31, wave = tid >> 5;
  const int lr = lane & 15, lh = lane >> 4;
  const int mt = wave >> 1;
  const int nb = (wave & 1) * 64;
  const int ar = tid >> 2, ac4 = (tid & 3) * 4;        // A-stage mapping (1 x b128/thread)
  const int br = tid >> 5, bc = (tid & 31) * 4;        // B-stage mapping (2 x float4/thread)

  float4 b0, b1;
  // prologue: stage chunk 0 into buffer 0
  async_load_b128(&As[0][ar][ac4], X + (size_t)(m0 + ar) * H_ + ac4);
  b0 = *reinterpret_cast<const float4*>(W + (size_t)br * H_ + n0 + bc);
  b1 = *reinterpret_cast<const float4*>(W + (size_t)(8 + br) * H_ + n0 + bc);
  BsT[0][bc + 0][br] = b0.x; BsT[0][bc + 1][br] = b0.y;
  BsT[0][bc + 2][br] = b0.z; BsT[0][bc + 3][br] = b0.w;
  BsT[0][bc + 0][br + 8] = b1.x; BsT[0][bc + 1][br + 8] = b1.y;
  BsT[0][bc + 2][br + 8] = b1.z; BsT[0][bc + 3][br + 8] = b1.w;
  wait_async0();
  __syncthreads();

  for (int i = 0; i < H_ / 16; ++i) {
    const int cur = i & 1, nxt = cur ^ 1;
    const int kb_next = (i + 1) * 16;
    if (i < H_ / 16 - 1) {
      async_load_b128(&As[nxt][ar][ac4], X + (size_t)(m0 + ar) * H_ + kb_next + ac4);
      b0 = *reinterpret_cast<const float4*>(W + (size_t)(kb_next + br) * H_ + n0 + bc);
      b1 = *reinterpret_cast<const float4*>(W + (size_t)(kb_next + 8 + br) * H_ + n0 + bc);
    }
#pragma unroll
    for (int kk = 0; kk < 4; ++kk) {
      const v2f a = *reinterpret_cast<const v2f*>(&As[cur][mt * 16 + lr][kk * 4 + lh * 2]);
#pragma unroll
      for (int t = 0; t < 4; ++t) {
        const v2f bb = *reinterpret_cast<const v2f*>(&BsT[cur][nb + t * 16 + lr][kk * 4 + lh * 2]);
        g.acc[t] = wmma4(a, bb, g.acc[t]);
      }
    }
    if (i < H_ / 16 - 1) {
      BsT[nxt][bc + 0][br] = b0.x; BsT[nxt][bc + 1][br] = b0.y;
      BsT[nxt][bc + 2][br] = b0.z; BsT[nxt][bc + 3][br] = b0.w;
      BsT[nxt][bc + 0][br + 8] = b1.x; BsT[nxt][bc + 1][br + 8] = b1.y;
      BsT[nxt][bc + 2][br + 8] = b1.z; BsT[nxt][bc + 3][br + 8] = b1.w;
      wait_async0();
      __syncthreads();
    }
  }
}

// ---------------- fused QKV projection GEMM ----------------
// C[2048,1024] = X @ W + b ; scatter to [b, nh, s, hd]. blockIdx.z selects Q/K/V.

__global__ __launch_bounds__(256) void qkv_gemm_kernel(
    const float* __restrict__ q_in, const float* __restrict__ k_in, const float* __restrict__ v_in,
    const float* __restrict__ Wq, const float* __restrict__ bq,
    const float* __restrict__ Wk, const float* __restrict__ bk,
    const float* __restrict__ Wv, const float* __restrict__ bv,
    float* __restrict__ Qo, float* __restrict__ Ko, float* __restrict__ Vo) {
  __shared__ float As[2][64][AS_STRIDE];
  __shared__ float BsT[2][128][BS_STRIDE];
  const float* X; const float* W; const float* bias; float* dst;
  if (blockIdx.z == 0)      { X = q_in; W = Wq; bias = bq; dst = Qo; }
  else if (blockIdx.z == 1) { X = k_in; W = Wk; bias = bk; dst = Ko; }
  else                      { X = v_in; W = Wv; bias = bv; dst = Vo; }
  const int m0 = blockIdx.x * 64, n0 = blockIdx.y * 128;
  const int lane = threadIdx.x & 31, wave = threadIdx.x >> 5;
  const int lr = lane & 15, lh = lane >> 4;
  const int mt = wave >> 1;
  const int nb = (wave & 1) * 64;
  GemmAcc g = {};
  gemm_mainloop(X, W, m0, n0, As, BsT, g);
#pragma unroll
  for (int t = 0; t < 4; ++t) {
    const int ncol = n0 + nb + t * 16 + lr;
    const float bvv = bias[ncol];
    const int head = ncol >> 6, hdc = ncol & 63;
#pragma unroll
    for (int r = 0; r < 8; ++r) {
      const int m = m0 + mt * 16 + lh * 8 + r;
      const int bidx = m >> 10, srow = m & (S_ - 1);
      dst[((size_t)(bidx * NH_ + head) * S_ + srow) * HD_ + hdc] = g.acc[t][r] + bvv;
    }
  }
}

// ---------------- output projection + bias + residual ----------------

__global__ __launch_bounds__(256) void oproj_gemm_kernel(
    const float* __restrict__ CTX, const float* __restrict__ Wo, const float* __restrict__ bo,
    const float* __restrict__ query, float* __restrict__ ATT) {
  __shared__ float As[2][64][AS_STRIDE];
  __shared__ float BsT[2][128][BS_STRIDE];
  const int m0 = blockIdx.x * 64, n0 = blockIdx.y * 128;
  const int lane = threadIdx.x & 31, wave = threadIdx.x >> 5;
  const int lr = lane & 15, lh = lane >> 4;
  const int mt = wave >> 1;
  const int nb = (wave & 1) * 64;
  GemmAcc g = {};
  gemm_mainloop(CTX, Wo, m0, n0, As, BsT, g);
#pragma unroll
  for (int t = 0; t < 4; ++t) {
    const int ncol = n0 + nb + t * 16 + lr;
    const float bvv = bo[ncol];
#pragma unroll
    for (int r = 0; r < 8; ++r) {
      const int m = m0 + mt * 16 + lh * 8 + r;
      ATT[(size_t)m * H_ + ncol] = g.acc[t][r] + bvv + query[(size_t)m * H_ + ncol];
    }
  }
}

// ---------------- flash-style gated attention (WMMA f32) ----------------
// Block: 128 threads = 4 waves; each wave owns 16 q-rows; block owns 64 q-rows of one (b,h).
// Ks [kcol][hd] (async-staged), VsT [hd][kcol] (transposed), both padded stride 68.

#define KS_STRIDE 68

__global__ __launch_bounds__(128) void attn_kernel(
    const float* __restrict__ Q, const float* __restrict__ K, const float* __restrict__ V,
    const float* __restrict__ msbh, const float* __restrict__ spec,
    float* __restrict__ CTX) {
  __shared__ float Ks[64][KS_STRIDE];
  __shared__ float VsT[64][KS_STRIDE];
  __shared__ float Ps[4][16][KS_STRIDE];  // per-wave P staging (C-layout -> A-layout)
  const int qb = blockIdx.x * 64;
  const int h = blockIdx.y, b = blockIdx.z;
  const size_t hb = (size_t)(b * NH_ + h) * S_ * HD_;
  const float* Qh = Q + hb; const float* Kh = K + hb; const float* Vh = V + hb;
  const int tid = threadIdx.x, lane = tid & 31, wave = tid >> 5;
  const int lr = lane & 15, lh = lane >> 4;

  // Q fragments in A-layout: lane lr = row, k = kk*4 + lh*2 + {0,1} (b64 global loads)
  v2f qf[16];
#pragma unroll
  for (int kk = 0; kk < 16; ++kk)
    qf[kk] = *reinterpret_cast<const v2f*>(
        Qh + (size_t)(qb + wave * 16 + lr) * HD_ + kk * 4 + lh * 2);

  const float mh = msbh[h];
  const float spb = spec[b];
  const float scale = 0.125f;  // 1/sqrt(64)

  float mrow[8], lrow[8];
#pragma unroll
  for (int r = 0; r < 8; ++r) { mrow[r] = -1e30f; lrow[r] = 0.f; }
  v8f oacc[4] = {};

  for (int kc = 0; kc < S_ / 64; ++kc) {
    __syncthreads();
    // K chunk: async b128 copies (contiguous per lane)
#pragma unroll
    for (int i = 0; i < 8; ++i) {
      const int idx = tid + i * 128;
      const int row = idx >> 4, c4 = (idx & 15) * 4;
      async_load_b128(&Ks[row][c4], Kh + (size_t)(kc * 64 + row) * HD_ + c4);
    }
    // V chunk: load + transpose into VsT[hd][kcol]
#pragma unroll
    for (int i = 0; i < 8; ++i) {
      const int idx = tid + i * 128;
      const int row = idx >> 4, c4 = (idx & 15) * 4;
      const float4 vv = *reinterpret_cast<const float4*>(Vh + (size_t)(kc * 64 + row) * HD_ + c4);
      VsT[c4 + 0][row] = vv.x; VsT[c4 + 1][row] = vv.y;
      VsT[c4 + 2][row] = vv.z; VsT[c4 + 3][row] = vv.w;
    }
    wait_async0();
    __syncthreads();

    // scores 16x64 per wave: S = Q Kt (B frags are contiguous b64 in Ks)
    v8f sacc[4] = {};
#pragma unroll
    for (int kk = 0; kk < 16; ++kk) {
      const v2f a = qf[kk];
#pragma unroll
      for (int t = 0; t < 4; ++t) {
        const v2f bb = *reinterpret_cast<const v2f*>(&Ks[t * 16 + lr][kk * 4 + lh * 2]);
        sacc[t] = wmma4(a, bb, sacc[t]);
      }
    }
    // gate: s' = s*scale; s' *= (1 + sigmoid(s'*msbh)*SP) * spec
#pragma unroll
    for (int t = 0; t < 4; ++t)
#pragma unroll
      for (int r = 0; r < 8; ++r) {
        const float s = sacc[t][r] * scale;
        const float e = 1.f / (1.f + __expf(-s * mh));
        sacc[t][r] = s * (1.f + e * SP_) * spb;
      }
    // online softmax: rows live in (VGPR r, 16-lane half)
    float scl[8];
#pragma unroll
    for (int r = 0; r < 8; ++r) {
      float rm = fmaxf(fmaxf(sacc[0][r], sacc[1][r]), fmaxf(sacc[2][r], sacc[3][r]));
      rm = fmaxf(rm, __shfl_xor(rm, 1, 32));
      rm = fmaxf(rm, __shfl_xor(rm, 2, 32));
      rm = fmaxf(rm, __shfl_xor(rm, 4, 32));
      rm = fmaxf(rm, __shfl_xor(rm, 8, 32));
      const float mnew = fmaxf(mrow[r], rm);
      scl[r] = __expf(mrow[r] - mnew);
      mrow[r] = mnew;
    }
#pragma unroll
    for (int t = 0; t < 4; ++t)
#pragma unroll
      for (int r = 0; r < 8; ++r)
        sacc[t][r] = __expf(sacc[t][r] - mrow[r]);
#pragma unroll
    for (int r = 0; r < 8; ++r) {
      float rs = sacc[0][r] + sacc[1][r] + sacc[2][r] + sacc[3][r];
      rs += __shfl_xor(rs, 1, 32);
      rs += __shfl_xor(rs, 2, 32);
      rs += __shfl_xor(rs, 4, 32);
      rs += __shfl_xor(rs, 8, 32);
      lrow[r] = lrow[r] * scl[r] + rs;
#pragma unroll
      for (int t = 0; t < 4; ++t) oacc[t][r] *= scl[r];
    }
    // stage P through per-wave LDS (C-layout positions; same-wave DS ops are in-order)
#pragma unroll
    for (int t = 0; t < 4; ++t)
#pragma unroll
      for (int r = 0; r < 8; ++r)
        Ps[wave][lh * 8 + r][t * 16 + lr] = sacc[t][r];
    // ctx += P @ Vchunk  (A frags contiguous b64 from Ps; B frags contiguous b64 from VsT)
#pragma unroll
    for (int kk = 0; kk < 16; ++kk) {
      const v2f a = *reinterpret_cast<const v2f*>(&Ps[wave][lr][kk * 4 + lh * 2]);
#pragma unroll
      for (int t = 0; t < 4; ++t) {
        const v2f bb = *reinterpret_cast<const v2f*>(&VsT[t * 16 + lr][kk * 4 + lh * 2]);
        oacc[t] = wmma4(a, bb, oacc[t]);
      }
    }
  }
  // finalize: divide by l, write ctx back to [b, s, H] (heads re-interleaved)
#pragma unroll
  for (int t = 0; t < 4; ++t) {
    const int col = h * HD_ + t * 16 + lr;
#pragma unroll
    for (int r = 0; r < 8; ++r) {
      const int srow = qb + wave * 16 + lh * 8 + r;
      CTX[(size_t)(b * S_ + srow) * H_ + col] = oacc[t][r] / lrow[r];
    }
  }
}

// ---------------- LayerNorm epilogue ----------------

__global__ __launch_bounds__(256) void ln_kernel(const float* __restrict__ ATT,
                                                 const float* __restrict__ g,
                                                 const float* __restrict__ beta,
                                                 float* __restrict__ out) {
  __shared__ float red[16];
  __shared__ float stat[2];
  const int row = blockIdx.x;
  const float* x = ATT + (size_t)row * H_;
  float v[4];
  float s = 0.f, ss = 0.f;
#pragma unroll
  for (int i = 0; i < 4; ++i) {
    v[i] = x[threadIdx.x + i * 256];
    s += v[i]; ss += v[i] * v[i];
  }
  for (int m = 16; m >= 1; m >>= 1) { s += __shfl_xor(s, m, 32); ss += __shfl_xor(ss, m, 32); }
  const int lane = threadIdx.x & 31, wave = threadIdx.x >> 5;
  if (lane == 0) { red[wave] = s; red[8 + wave] = ss; }
  __syncthreads();
  if (threadIdx.x == 0) {
    float ts = 0.f, tss = 0.f;
    for (int w = 0; w < 8; ++w) { ts += red[w]; tss += red[8 + w]; }
    const float mean = ts / (float)H_;
    const float var = tss / (float)H_ - mean * mean;
    stat[0] = mean;
    stat[1] = rsqrtf(var + EPS_);
  }
  __syncthreads();
  const float mean = stat[0], rstd = stat[1];
#pragma unroll
  for (int i = 0; i < 4; ++i) {
    const int c = threadIdx.x + i * 256;
    out[(size_t)row * H_ + c] = (v[i] - mean) * rstd * g[c] + beta[c];
  }
}

// ---------------- host-side launch ----------------

extern "C" void kernel_launch(void* const* d_in, const int* in_sizes, int n_in,
                              void* d_out, int out_size, void* d_ws, size_t ws_size,
                              hipStream_t stream) {
  (void)in_sizes; (void)n_in; (void)out_size; (void)ws_size;
  const float* query = (const float*)d_in[0];
  const float* key_t = (const float*)d_in[1];
  const float* value = (const float*)d_in[2];
  const float* Wq = (const float*)d_in[3];  const float* bq = (const float*)d_in[4];
  const float* Wk = (const float*)d_in[5];  const float* bk = (const float*)d_in[6];
  const float* Wv = (const float*)d_in[7];  const float* bv = (const float*)d_in[8];
  const float* msb = (const float*)d_in[9];
  const float* Ws1 = (const float*)d_in[10]; const float* bs1 = (const float*)d_in[11];
  const float* Ws2 = (const float*)d_in[12]; const float* bs2 = (const float*)d_in[13];
  const float* Wo = (const float*)d_in[14];  const float* bo = (const float*)d_in[15];
  const float* ln_g = (const float*)d_in[16];
  const float* ln_b = (const float*)d_in[17];

  float* ws = (float*)d_ws;
  const size_t NBH = (size_t)B_ * S_ * H_;           // 2,097,152
  float* Qb     = ws;
  float* Kb     = ws + NBH;
  float* Vb     = ws + 2 * NBH;
  float* CTX    = ws + 3 * NBH;
  float* ATT    = ws + 4 * NBH;
  float* MSBH   = ws + 5 * NBH;
  float* SPECIN = MSBH + 16;
  float* SPEC   = SPECIN + (size_t)B_ * H_;

  msb_mean_kernel<<<NH_, 256, 0, stream>>>(msb, MSBH);
  spec_in_kernel<<<(B_ * H_) / 256, 256, 0, stream>>>(query, SPECIN);
  spec_mlp_kernel<<<B_, 256, 0, stream>>>(SPECIN, Ws1, bs1, Ws2, bs2, SPEC);
  qkv_gemm_kernel<<<dim3((B_ * S_) / 64, H_ / 128, 3), 256, 0, stream>>>(
      query, key_t, value, Wq, bq, Wk, bk, Wv, bv, Qb, Kb, Vb);
  attn_kernel<<<dim3(S_ / 64, NH_, B_), 128, 0, stream>>>(Qb, Kb, Vb, MSBH, SPEC, CTX);
  oproj_gemm_kernel<<<dim3((B_ * S_) / 64, H_ / 128), 256, 0, stream>>>(CTX, Wo, bo, query, ATT);
  ln_kernel<<<B_ * S_, 256, 0, stream>>>(ATT, ln_g, ln_b, (float*)d_out);
}